// Udf_loss_24773371363369
// MI455X (gfx1250) — compile-verified
//
#include <hip/hip_runtime.h>
#include <hip/hip_bf16.h>
#include <math.h>

typedef __attribute__((ext_vector_type(2))) float v2f;
typedef __attribute__((ext_vector_type(8))) float v8f;

#define D_DIM 64
#define J_DIM 20
#define C_DIM 3
#define VK (J_DIM * C_DIM * C_DIM) /* 180 = 9 batches x 5 steps x K4 */

#define LD2(p) (*(const v2f*)(p))

// ---------------------------------------------------------------------------
// Kernel 0: zero the scalar accumulator (d_out is poisoned by the harness).
// ---------------------------------------------------------------------------
__global__ void zero_out_kernel(float* out) { out[0] = 0.0f; }

// ---------------------------------------------------------------------------
// Kernel 1: per-row preprocessing.
//   sqn[i] = sum_d predictions[i,d]^2
//   V[i,:] = concat over 20 joints of vec( u u^T ), u = joint / max(|joint|,eps)
// so that sum_k cos^2(i,j,k) == dot(V[i,:], V[j,:])  (a K=180 f32 GEMM).
// ---------------------------------------------------------------------------
__global__ void prep_kernel(const float* __restrict__ preds,
                            const float* __restrict__ data,
                            float* __restrict__ V,
                            float* __restrict__ sqn, int N) {
  int i = blockIdx.x * blockDim.x + threadIdx.x;
  if (i >= N) return;

  const float* p = preds + (size_t)i * D_DIM;
  float s = 0.0f;
#pragma unroll
  for (int d = 0; d < D_DIM; ++d) s = fmaf(p[d], p[d], s);
  sqn[i] = s;

  const float* dr = data + (size_t)i * (J_DIM * C_DIM);
  float* vr = V + (size_t)i * VK;
#pragma unroll
  for (int k = 0; k < J_DIM; ++k) {
    float x = dr[3 * k + 0], y = dr[3 * k + 1], z = dr[3 * k + 2];
    float n = sqrtf(x * x + y * y + z * z);
    float inv = 1.0f / fmaxf(n, 1e-8f);
    float u[3] = {x * inv, y * inv, z * inv};
#pragma unroll
    for (int a = 0; a < 3; ++a)
#pragma unroll
      for (int b = 0; b < 3; ++b)
        vr[k * 9 + a * 3 + b] = u[a] * u[b];
  }
}

// ---------------------------------------------------------------------------
// Double-buffered 2x2-register-tile GEMM stage.
//   T batches x B K4-steps; total K = T*B*4.
//   Per K-step: 4 fragment loads feed 4 independent WMMAs (2x reuse of A & B).
//   Batch t+1's loads are issued as one clause BEFORE batch t's WMMAs, so the
//   s_wait for them overlaps 4*B back-to-back v_wmma_f32_16x16x4_f32.
// Fragment layout (ISA 7.12.2, 32-bit A 16x4 + its B mirror): lane L supplies
// a float2 from row (base + L%16) at col k0 + 2*(L>>4); pointers below are
// pre-offset by that lane term, so each step is just  base + 4*step.
// ---------------------------------------------------------------------------
template <int T, int B>
__device__ __forceinline__ void gemm2x2(const float* __restrict__ pi0,
                                        const float* __restrict__ pi1,
                                        const float* __restrict__ pj0,
                                        const float* __restrict__ pj1,
                                        v8f& c00, v8f& c01, v8f& c10, v8f& c11) {
  v2f A0[B], A1[B], B0[B], B1[B];
  v2f nA0[B], nA1[B], nB0[B], nB1[B];

#pragma unroll
  for (int s = 0; s < B; ++s) {
    A0[s] = LD2(pi0 + 4 * s);
    A1[s] = LD2(pi1 + 4 * s);
    B0[s] = LD2(pj0 + 4 * s);
    B1[s] = LD2(pj1 + 4 * s);
  }

#pragma unroll
  for (int t = 0; t < T; ++t) {
    if (t + 1 < T) {
      const int o = (t + 1) * B * 4;
#pragma unroll
      for (int s = 0; s < B; ++s) {
        nA0[s] = LD2(pi0 + o + 4 * s);
        nA1[s] = LD2(pi1 + o + 4 * s);
        nB0[s] = LD2(pj0 + o + 4 * s);
        nB1[s] = LD2(pj1 + o + 4 * s);
      }
    }
#pragma unroll
    for (int s = 0; s < B; ++s) {
      c00 = __builtin_amdgcn_wmma_f32_16x16x4_f32(false, A0[s], false, B0[s],
                                                  (short)0, c00, false, false);
      c01 = __builtin_amdgcn_wmma_f32_16x16x4_f32(false, A0[s], false, B1[s],
                                                  (short)0, c01, false, false);
      c10 = __builtin_amdgcn_wmma_f32_16x16x4_f32(false, A1[s], false, B0[s],
                                                  (short)0, c10, false, false);
      c11 = __builtin_amdgcn_wmma_f32_16x16x4_f32(false, A1[s], false, B1[s],
                                                  (short)0, c11, false, false);
    }
    if (t + 1 < T) {
#pragma unroll
      for (int s = 0; s < B; ++s) {
        A0[s] = nA0[s]; A1[s] = nA1[s]; B0[s] = nB0[s]; B1[s] = nB1[s];
      }
    }
  }
}

// ---------------------------------------------------------------------------
// Per-16x16-subtile epilogue: dist/dis from the two accumulators, masked sum.
// C/D layout: n = lane%16, m = vgpr + 8*(lane>=16).
// ---------------------------------------------------------------------------
__device__ __forceinline__ float tile_epilogue(const v8f& cg, const v8f& cv,
                                               const float* __restrict__ sqn,
                                               int ti, int tj, int m16, int half) {
  const int gj = tj * 16 + m16;
  const float qj = sqn[gj];
  float acc = 0.0f;
#pragma unroll
  for (int r = 0; r < 8; ++r) {
    int gi = ti * 16 + r + half;
    float d2   = sqn[gi] + qj - 2.0f * cg[r];
    float dist = sqrtf(fmaxf(d2, 0.0f));
    float dis  = sqrtf(fmaxf((float)J_DIM - cv[r], 0.0f));
    acc += (gi < gj) ? fabsf(dist - dis) : 0.0f;  // strict upper triangle
  }
  return acc;
}

// ---------------------------------------------------------------------------
// Kernel 2: one wave32 per 32x32 block of the upper triangle (2x2 tiles).
// ---------------------------------------------------------------------------
__global__ __launch_bounds__(32)
void pair_loss_kernel(const float* __restrict__ P,
                      const float* __restrict__ V,
                      const float* __restrict__ sqn,
                      float* __restrict__ out,
                      float scale) {
  const int bi = blockIdx.x;
  const int bj = blockIdx.y;
  if (bj < bi) return;  // uniform wave exit: EXEC stays all-1s for WMMA

  const int lane = threadIdx.x;
  const int m16  = lane & 15;
  const int kh   = (lane >> 4) << 1;
  const int half = (lane >> 4) * 8;

  // --- Gram GEMM over predictions (K = 64 = 4 batches x 4 steps) ---
  const float* Pi0 = P + (size_t)(bi * 32 + m16) * D_DIM + kh;
  const float* Pi1 = Pi0 + 16 * D_DIM;
  const float* Pj0 = P + (size_t)(bj * 32 + m16) * D_DIM + kh;
  const float* Pj1 = Pj0 + 16 * D_DIM;
  v8f g00 = {}, g01 = {}, g10 = {}, g11 = {};
  gemm2x2<4, 4>(Pi0, Pi1, Pj0, Pj1, g00, g01, g10, g11);

  // --- cos^2 GEMM over V features (K = 180 = 9 batches x 5 steps) ---
  const float* Vi0 = V + (size_t)(bi * 32 + m16) * VK + kh;
  const float* Vi1 = Vi0 + 16 * VK;
  const float* Vj0 = V + (size_t)(bj * 32 + m16) * VK + kh;
  const float* Vj1 = Vj0 + 16 * VK;
  v8f v00 = {}, v01 = {}, v10 = {}, v11 = {};
  gemm2x2<9, 5>(Vi0, Vi1, Vj0, Vj1, v00, v01, v10, v11);

  // --- epilogue over the 4 sub-tiles ---
  const int ti0 = bi * 2, ti1 = bi * 2 + 1;
  const int tj0 = bj * 2, tj1 = bj * 2 + 1;
  float acc = 0.0f;
  acc += tile_epilogue(g00, v00, sqn, ti0, tj0, m16, half);
  acc += tile_epilogue(g01, v01, sqn, ti0, tj1, m16, half);
  acc += tile_epilogue(g10, v10, sqn, ti1, tj0, m16, half);
  acc += tile_epilogue(g11, v11, sqn, ti1, tj1, m16, half);

  // wave32 reduction, then one scaled atomic per block
  for (int off = 16; off > 0; off >>= 1)
    acc += __shfl_xor(acc, off, 32);
  if (lane == 0) atomicAdd(out, acc * scale);
}

// ---------------------------------------------------------------------------
extern "C" void kernel_launch(void* const* d_in, const int* in_sizes, int n_in,
                              void* d_out, int out_size, void* d_ws, size_t ws_size,
                              hipStream_t stream) {
  const float* preds = (const float*)d_in[0];   // [N, 64]
  const float* data  = (const float*)d_in[1];   // [N, 20, 3]
  float* out = (float*)d_out;                   // scalar f32

  const int N = in_sizes[0] / D_DIM;            // 2048

  float* V   = (float*)d_ws;                    // [N, 180]
  float* sqn = V + (size_t)N * VK;              // [N]

  const long long num = (long long)N * (N + 1) / 2 - 1;
  const float scale = 1.0f / (float)num;

  zero_out_kernel<<<1, 1, 0, stream>>>(out);
  prep_kernel<<<(N + 255) / 256, 256, 0, stream>>>(preds, data, V, sqn, N);

  const int nb = N / 32;                        // 64 (N is a multiple of 32)
  pair_loss_kernel<<<dim3(nb, nb), 32, 0, stream>>>(preds, V, sqn, out, scale);
}